// LATTE_11132555231329
// MI455X (gfx1250) — compile-verified
//
#include <hip/hip_runtime.h>
#include <math.h>
#include <stdint.h>

#define NN   100000
#define EE   1000000
#define DINN 256
#define DDIM 128
#define HHH  4
#define CCC  32
#define NEG_SLOPE 0.2f
#define LSTRIDE 260   // 256 + 4 pad: rows hit distinct LDS banks; row stride 1040B (16B aligned)

typedef __attribute__((ext_vector_type(2))) float v2f;
typedef __attribute__((ext_vector_type(8))) float v8f;

#if __has_builtin(__builtin_amdgcn_global_load_async_to_lds_b128)
#define HAVE_ASYNC_LDS 1
typedef int v4i __attribute__((vector_size(16)));
typedef __attribute__((address_space(1))) v4i g_v4i;   // global 128-bit chunk
typedef __attribute__((address_space(3))) v4i l_v4i;   // LDS 128-bit chunk
// int->pointer reinterpret is valid for any addrspace; generic shared flat
// address low 32 bits == LDS byte offset (flat aperture rule, ISA 10.2)
#define TO_GLB(p) ((g_v4i*)(uintptr_t)(p))
#define TO_LDS(p) ((l_v4i*)(unsigned)(uintptr_t)(p))
#else
#define HAVE_ASYNC_LDS 0
#endif

__device__ __forceinline__ void wait_asynccnt0() {
#if __has_builtin(__builtin_amdgcn_s_wait_asynccnt)
    __builtin_amdgcn_s_wait_asynccnt(0);
#else
    asm volatile("s_wait_asynccnt 0" ::: "memory");
#endif
}

__device__ __forceinline__ unsigned fenc(float f) {
    unsigned u = __float_as_uint(f);
    return (u & 0x80000000u) ? ~u : (u | 0x80000000u);
}
__device__ __forceinline__ float fdec(unsigned e) {
    unsigned u = (e & 0x80000000u) ? (e ^ 0x80000000u) : ~e;
    return __uint_as_float(u);
}
__device__ __forceinline__ float lrelu(float x) { return x > 0.f ? x : NEG_SLOPE * x; }

// ---------------------------------------------------------------------------
// Kernel A: fused linear_l / linear_r via V_WMMA_F32_16X16X4_F32 + beta softmax
// block = 16 nodes, 8 waves; wave w owns output dims [16w, 16w+16) for BOTH GEMMs
// x tiles staged into LDS via async global->LDS (ASYNCcnt) when available
// ---------------------------------------------------------------------------
__global__ __launch_bounds__(256) void k_linear(
    const float* __restrict__ xl, const float* __restrict__ xr,
    const float* __restrict__ Wl, const float* __restrict__ bl,
    const float* __restrict__ Wr, const float* __restrict__ br,
    const float* __restrict__ cw, const float* __restrict__ cb,
    float* __restrict__ lout, float* __restrict__ rout, float* __restrict__ beta)
{
    __shared__ __align__(16) float sxl[16 * LSTRIDE];
    __shared__ __align__(16) float sxr[16 * LSTRIDE];
    const int nb  = blockIdx.x * 16;
    const int tid = threadIdx.x;

    // stage 16x256 f32 tiles of x_l / x_r (16B-aligned rows in LDS and global)
    for (int i = tid; i < 16 * 64; i += 256) {          // 1024 float4 per matrix
        const int row = i >> 6, c4 = i & 63;
        float*       dl = &sxl[row * LSTRIDE + c4 * 4];
        float*       dr = &sxr[row * LSTRIDE + c4 * 4];
        const float* gl = xl + (size_t)(nb + row) * DINN + c4 * 4;
        const float* gr = xr + (size_t)(nb + row) * DINN + c4 * 4;
#if HAVE_ASYNC_LDS
        __builtin_amdgcn_global_load_async_to_lds_b128(TO_GLB(gl), TO_LDS(dl), 0, 0);
        __builtin_amdgcn_global_load_async_to_lds_b128(TO_GLB(gr), TO_LDS(dr), 0, 0);
#else
        *(float4*)dl = *(const float4*)gl;
        *(float4*)dr = *(const float4*)gr;
#endif
    }
#if HAVE_ASYNC_LDS
    wait_asynccnt0();
#endif
    __syncthreads();

    const int lane = tid & 31;
    const int wave = tid >> 5;           // N-tile (0..7), D = 128 = 8*16
    const int half = lane >> 4;          // 0: K=0,1 ; 1: K=2,3   (A-matrix layout)
    const int mrow = lane & 15;          // A-matrix M row / B-matrix N col
    const int dim  = wave * 16 + mrow;   // output feature index
    const float* wlrow = Wl + (size_t)dim * DINN;   // Wl[d][k]
    const float* wrrow = Wr + (size_t)dim * DINN;

    v8f accl = {}; v8f accr = {};
#if __has_builtin(__builtin_amdgcn_wmma_f32_16x16x4_f32)
    #pragma unroll 8
    for (int k0 = 0; k0 < DINN; k0 += 4) {
        const int kk = k0 + 2 * half;
        v2f a, b;
        a.x = sxl[mrow * LSTRIDE + kk]; a.y = sxl[mrow * LSTRIDE + kk + 1];
        b.x = wlrow[kk];                b.y = wlrow[kk + 1];
        accl = __builtin_amdgcn_wmma_f32_16x16x4_f32(false, a, false, b,
                                                     (short)0, accl, false, false);
        a.x = sxr[mrow * LSTRIDE + kk]; a.y = sxr[mrow * LSTRIDE + kk + 1];
        b.x = wrrow[kk];                b.y = wrrow[kk + 1];
        accr = __builtin_amdgcn_wmma_f32_16x16x4_f32(false, a, false, b,
                                                     (short)0, accr, false, false);
    }
#else
    for (int v = 0; v < 8; ++v) {
        int rrow = v + 8 * half;
        float s = 0.f, t = 0.f;
        for (int k = 0; k < DINN; ++k) {
            s += sxl[rrow * LSTRIDE + k] * wlrow[k];
            t += sxr[rrow * LSTRIDE + k] * wrrow[k];
        }
        accl[v] = s; accr[v] = t;
    }
#endif
    const float bld = bl[dim], brd = br[dim];
    #pragma unroll
    for (int v = 0; v < 8; ++v) {            // C/D layout: VGPR v -> row v (+8 upper half)
        int node = nb + v + 8 * half;
        lout[(size_t)node * DDIM + dim] = accl[v] + bld;
        rout[(size_t)node * DDIM + dim] = accr[v] + brd;
    }

    // beta = softmax(x_r @ conv_w.T + conv_b) over R=3, one thread per node
    if (tid < 16) {
        float a0 = cb[0], a1 = cb[1], a2 = cb[2];
        const float* xrow = &sxr[tid * LSTRIDE];
        for (int k = 0; k < DINN; ++k) {
            float xv = xrow[k];
            a0 += xv * cw[k];
            a1 += xv * cw[DINN + k];
            a2 += xv * cw[2 * DINN + k];
        }
        float mx = fmaxf(a0, fmaxf(a1, a2));
        float e0 = __expf(a0 - mx), e1 = __expf(a1 - mx), e2 = __expf(a2 - mx);
        float inv = 1.f / (e0 + e1 + e2);
        int n = nb + tid;
        beta[n * 3 + 0] = e0 * inv;
        beta[n * 3 + 1] = e1 * inv;
        beta[n * 3 + 2] = e2 * inv;
    }
}

// ---------------------------------------------------------------------------
// Kernel B: per-node attention scores  s_i[m][n,h], s_j[m][n,h]
// ---------------------------------------------------------------------------
__global__ __launch_bounds__(256) void k_scores(
    const float* __restrict__ l, const float* __restrict__ r,
    const float* __restrict__ attn,
    float* __restrict__ si, float* __restrict__ sj)
{
    int idx = blockIdx.x * 256 + threadIdx.x;
    if (idx >= NN * HHH) return;
    int n = idx >> 2, h = idx & 3;
    const float* rp = r + (size_t)n * DDIM + h * CCC;
    const float* lp = l + (size_t)n * DDIM + h * CCC;
    #pragma unroll
    for (int m = 0; m < 2; ++m) {
        const float* am = attn + (size_t)(m * HHH + h) * (2 * CCC);
        float a = 0.f, b = 0.f;
        #pragma unroll 8
        for (int c = 0; c < CCC; ++c) {
            a += lrelu(rp[c]) * am[c];
            b += lrelu(lp[c]) * am[CCC + c];
        }
        si[(size_t)m * NN * HHH + idx] = a;
        sj[(size_t)m * NN * HHH + idx] = b;
    }
}

// ---------------------------------------------------------------------------
// Scratch zeroing (accumulators / max / sum) — runs every call for determinism
// ---------------------------------------------------------------------------
__global__ void k_zero(float* __restrict__ p, long long n) {
    long long i = (long long)blockIdx.x * blockDim.x + threadIdx.x;
    long long stride = (long long)gridDim.x * blockDim.x;
    for (; i < n; i += stride) p[i] = 0.f;   // bits 0 == fenc identity for max
}

// ---------------------------------------------------------------------------
// Edge pass 1: alpha = sharp*(s_i[dst]+s_j[src]); per-dst atomic max (ordered-uint)
// ---------------------------------------------------------------------------
__global__ __launch_bounds__(256) void k_edge_logit(
    const int* __restrict__ ei, const float* __restrict__ si,
    const float* __restrict__ sj, const float* __restrict__ sharp,
    float* __restrict__ alpha, unsigned* __restrict__ mx)
{
    int idx = blockIdx.x * 256 + threadIdx.x;
    if (idx >= EE * HHH) return;
    int e = idx >> 2, h = idx & 3;
    int src = ei[e], dst = ei[EE + e];
    float a = sharp[0] * (si[dst * HHH + h] + sj[src * HHH + h]);
    alpha[idx] = a;
    atomicMax(&mx[dst * HHH + h], fenc(a));
}

// ---------------------------------------------------------------------------
// Edge pass 2: e = exp(alpha - max[dst]); per-dst atomic sum
// ---------------------------------------------------------------------------
__global__ __launch_bounds__(256) void k_edge_exp(
    const int* __restrict__ ei, const unsigned* __restrict__ mx,
    float* __restrict__ alpha, float* __restrict__ sum)
{
    int idx = blockIdx.x * 256 + threadIdx.x;
    if (idx >= EE * HHH) return;
    int e = idx >> 2, h = idx & 3;
    int dst = ei[EE + e];
    float m = fdec(mx[dst * HHH + h]);
    if (!(fabsf(m) < 3.4e38f)) m = 0.f;       // non-finite segment max -> 0
    float ev = __expf(alpha[idx] - m);
    alpha[idx] = ev;
    atomicAdd(&sum[dst * HHH + h], ev);
}

// ---------------------------------------------------------------------------
// Edge pass 3: scatter  out[dst] += l[src] * e/(sum[dst]+eps)   (one thread per edge-head, loops all 32 of C)
// ---------------------------------------------------------------------------
__global__ __launch_bounds__(256) void k_edge_scatter(
    const int* __restrict__ ei, const float* __restrict__ l,
    const float* __restrict__ alpha, const float* __restrict__ sum,
    float* __restrict__ out)
{
    int idx = blockIdx.x * 256 + threadIdx.x;
    if (idx >= EE * HHH) return;
    int e = idx >> 2, h = idx & 3;
    int src = ei[e], dst = ei[EE + e];
    float w = alpha[idx] / (sum[dst * HHH + h] + 1e-16f);
    const float4* lp = (const float4*)(l + (size_t)src * DDIM + h * CCC);
    float* op = out + (size_t)dst * DDIM + h * CCC;
    #pragma unroll
    for (int c4 = 0; c4 < CCC / 4; ++c4) {
        float4 v = lp[c4];
        atomicAdd(&op[c4 * 4 + 0], v.x * w);
        atomicAdd(&op[c4 * 4 + 1], v.y * w);
        atomicAdd(&op[c4 * 4 + 2], v.z * w);
        atomicAdd(&op[c4 * 4 + 3], v.w * w);
    }
}

// ---------------------------------------------------------------------------
// Final blend: out = relu(b0*out0 + b1*out1 + b2*r)
// ---------------------------------------------------------------------------
__global__ __launch_bounds__(256) void k_combine(
    const float* __restrict__ out0, const float* __restrict__ out1,
    const float* __restrict__ r, const float* __restrict__ beta,
    float* __restrict__ out)
{
    int idx = blockIdx.x * 256 + threadIdx.x;
    if (idx >= NN * DDIM) return;
    int n = idx >> 7;
    float v = beta[n * 3 + 0] * out0[idx]
            + beta[n * 3 + 1] * out1[idx]
            + beta[n * 3 + 2] * r[idx];
    out[idx] = fmaxf(v, 0.f);
}

extern "C" void kernel_launch(void* const* d_in, const int* in_sizes, int n_in,
                              void* d_out, int out_size, void* d_ws, size_t ws_size,
                              hipStream_t stream)
{
    (void)in_sizes; (void)n_in; (void)out_size; (void)ws_size;
    const float* xl    = (const float*)d_in[0];
    const float* xr    = (const float*)d_in[1];
    const float* Wl    = (const float*)d_in[2];
    const float* bl    = (const float*)d_in[3];
    const float* Wr    = (const float*)d_in[4];
    const float* br    = (const float*)d_in[5];
    const float* cw    = (const float*)d_in[6];
    const float* cb    = (const float*)d_in[7];
    const float* attn  = (const float*)d_in[8];
    const float* sharp = (const float*)d_in[9];
    const int*   ei0   = (const int*)d_in[10];
    const int*   ei1   = (const int*)d_in[11];
    float* out = (float*)d_out;

    // workspace carve-out (floats); total ~251 MB
    float* ws     = (float*)d_ws;
    float* l      = ws;                                  // N*D
    float* r      = l      + (size_t)NN * DDIM;          // N*D
    float* beta   = r      + (size_t)NN * DDIM;          // N*3
    float* si     = beta   + (size_t)NN * 3;             // 2*N*H
    float* sj     = si     + (size_t)2 * NN * HHH;       // 2*N*H
    float* alpha0 = sj     + (size_t)2 * NN * HHH;       // E*H
    float* alpha1 = alpha0 + (size_t)EE * HHH;           // E*H
    float* zbase  = alpha1 + (size_t)EE * HHH;           // zero-initialized region:
    unsigned* mx0 = (unsigned*)zbase;                    //   N*H
    unsigned* mx1 = mx0    + (size_t)NN * HHH;           //   N*H
    float* sum0   = (float*)(mx1 + (size_t)NN * HHH);    //   N*H
    float* sum1   = sum0   + (size_t)NN * HHH;           //   N*H
    float* out0   = sum1   + (size_t)NN * HHH;           //   N*D
    float* out1   = out0   + (size_t)NN * DDIM;          //   N*D
    const long long zcount = (long long)4 * NN * HHH + (long long)2 * NN * DDIM;

    k_zero<<<2048, 256, 0, stream>>>(zbase, zcount);

    k_linear<<<NN / 16, 256, 0, stream>>>(xl, xr, Wl, bl, Wr, br, cw, cb, l, r, beta);

    k_scores<<<(NN * HHH + 255) / 256, 256, 0, stream>>>(l, r, attn, si, sj);

    const int egrid = (EE * HHH + 255) / 256;
    // metapath 0
    k_edge_logit  <<<egrid, 256, 0, stream>>>(ei0, si, sj, sharp, alpha0, mx0);
    k_edge_exp    <<<egrid, 256, 0, stream>>>(ei0, mx0, alpha0, sum0);
    k_edge_scatter<<<egrid, 256, 0, stream>>>(ei0, l, alpha0, sum0, out0);
    // metapath 1
    k_edge_logit  <<<egrid, 256, 0, stream>>>(ei1, si + (size_t)NN * HHH,
                                              sj + (size_t)NN * HHH, sharp + 1,
                                              alpha1, mx1);
    k_edge_exp    <<<egrid, 256, 0, stream>>>(ei1, mx1, alpha1, sum1);
    k_edge_scatter<<<egrid, 256, 0, stream>>>(ei1, l, alpha1, sum1, out1);

    k_combine<<<(NN * DDIM + 255) / 256, 256, 0, stream>>>(out0, out1, r, beta, out);
}